// MultiHeadDotGAT_9878424781458
// MI455X (gfx1250) — compile-verified
//
#include <hip/hip_runtime.h>
#include <math.h>

// Problem constants (from reference): N=20000, D_IN=128, D_HID=128, H=4,
// FINAL=H*D_HID=512, D_OUT=128, E=320000. All divisible by tile sizes.
#define N_NODES 20000
#define D_IN    128
#define FINAL   512
#define H_HEADS 4
#define D_OUT_  128
#define NEDGE   320000

typedef __attribute__((ext_vector_type(2))) float v2f;
typedef __attribute__((ext_vector_type(8))) float v8f;

// Monotone float<->uint key for atomicMax-based float max.
__device__ __forceinline__ unsigned fkey(float f) {
  unsigned u = __float_as_uint(f);
  return (u & 0x80000000u) ? ~u : (u | 0x80000000u);
}
__device__ __forceinline__ float funkey(unsigned k) {
  return (k & 0x80000000u) ? __uint_as_float(k ^ 0x80000000u)
                           : __uint_as_float(~k);
}

// ---------------------------------------------------------------------------
// Kernel 1: Q/K/R = x @ W^T (+bias). W flat (512,128). Output (N,512).
// One block = 16 rows; 8 waves * 4 col-tiles = 32 tiles of 16 cols.
// WMMA f32 16x16x4, A staged through LDS.
// ---------------------------------------------------------------------------
__global__ void k_gemm_qkr(const float* __restrict__ x,
                           const float* __restrict__ Wq,
                           const float* __restrict__ Wk,
                           const float* __restrict__ Wr,
                           const float* __restrict__ br,
                           float* __restrict__ Q, float* __restrict__ K,
                           float* __restrict__ R) {
  __shared__ float xs[16 * D_IN]; // 8 KB
  const int r0 = blockIdx.x * 16;
  const int which = blockIdx.y;
  const float* W;
  float* out;
  const float* bias = nullptr;
  if (which == 0)      { W = Wq; out = Q; }
  else if (which == 1) { W = Wk; out = K; }
  else                 { W = Wr; out = R; bias = br; }

  // Stage x tile [16][128] (coalesced float4 loads).
  {
    const int t = threadIdx.x;
#pragma unroll
    for (int j = 0; j < 2; ++j) {
      const int li = (t * 2 + j) * 4;          // float index 0..2047
      const int row = li >> 7, col = li & 127;
      *(float4*)&xs[li] = *(const float4*)&x[(size_t)(r0 + row) * D_IN + col];
    }
  }
  __syncthreads();

  const int wave = threadIdx.x >> 5;
  const int lane = threadIdx.x & 31;
  const int mn = lane & 15;      // A: row m; B/D: col n (same lane group)
  const int half = lane >> 4;    // K-pair select for A/B; +8 row offset for D

  for (int tile = 0; tile < 4; ++tile) {
    const int c0 = (wave + 8 * tile) * 16;
    v8f acc = {};
#pragma unroll 8
    for (int k0 = 0; k0 < D_IN; k0 += 4) {
      // A 16x4 fragment: lane holds X[r0+m][k0+2*half .. +1]
      v2f a = *(const v2f*)&xs[mn * D_IN + k0 + 2 * half];
      // B 4x16 fragment (B[k][n] = W[c0+n][k]): lane holds W row c0+n, k pair
      v2f b = *(const v2f*)&W[(size_t)(c0 + mn) * D_IN + k0 + 2 * half];
      acc = __builtin_amdgcn_wmma_f32_16x16x4_f32(
          false, a, false, b, (short)0, acc, false, false);
    }
    const float bv = bias ? bias[c0 + mn] : 0.0f;
#pragma unroll
    for (int r = 0; r < 8; ++r) {
      const int m = r + 8 * half; // D layout: lanes 16-31 hold rows M+8
      out[(size_t)(r0 + m) * FINAL + c0 + mn] = acc[r] + bv;
    }
  }
}

// ---------------------------------------------------------------------------
// Kernel 2: per-edge logits + global per-head max (for softmax stability).
// One wave per edge; lane loads float4, wave shuffle-reduce.
// ---------------------------------------------------------------------------
__global__ void k_edge_logits(const float* __restrict__ Q,
                              const float* __restrict__ K,
                              const int* __restrict__ erow,
                              const int* __restrict__ ecol,
                              float* __restrict__ LOG,
                              unsigned* __restrict__ MAXK) {
  __shared__ unsigned lk[H_HEADS];
  const int t = threadIdx.x;
  if (t < H_HEADS) lk[t] = 0u;
  __syncthreads();

  const int wave = t >> 5, lane = t & 31;
  const int e = blockIdx.x * 8 + wave;
  const int r = erow[e], c = ecol[e];
  const float4* qb = (const float4*)(Q + (size_t)r * FINAL);
  const float4* kb = (const float4*)(K + (size_t)c * FINAL);
  const float inv_scale = 0.0883883476483184f; // 1/sqrt(128)

#pragma unroll
  for (int h = 0; h < H_HEADS; ++h) {
    const float4 q4 = qb[h * 32 + lane];
    const float4 k4 = kb[h * 32 + lane];
    float acc = q4.x * k4.x + q4.y * k4.y + q4.z * k4.z + q4.w * k4.w;
#pragma unroll
    for (int off = 16; off > 0; off >>= 1) acc += __shfl_xor(acc, off, 32);
    if (lane == 0) {
      const float lg = acc * inv_scale;
      LOG[(size_t)h * NEDGE + e] = lg;
      atomicMax(&lk[h], fkey(lg));
    }
  }
  __syncthreads();
  if (t < H_HEADS) atomicMax(&MAXK[t], lk[t]);
}

// ---------------------------------------------------------------------------
// Kernel 3: exp(logit - max) in place + per-head global sum.
// ---------------------------------------------------------------------------
__global__ void k_exp_sum(float* __restrict__ LOG,
                          const unsigned* __restrict__ MAXK,
                          float* __restrict__ SUM) {
  __shared__ float ls[H_HEADS];
  const int t = threadIdx.x;
  if (t < H_HEADS) ls[t] = 0.0f;
  __syncthreads();
  const int e = blockIdx.x * 256 + t;
#pragma unroll
  for (int h = 0; h < H_HEADS; ++h) {
    const float mx = funkey(MAXK[h]);
    const float ex = __expf(LOG[(size_t)h * NEDGE + e] - mx);
    LOG[(size_t)h * NEDGE + e] = ex;
    atomicAdd(&ls[h], ex);
  }
  __syncthreads();
  if (t < H_HEADS) unsafeAtomicAdd(&SUM[t], ls[t]);
}

// ---------------------------------------------------------------------------
// Kernel 4: segment_sum scatter. One wave per edge; native f32 atomics.
// ---------------------------------------------------------------------------
__global__ void k_scatter(const float* __restrict__ Q,
                          const float* __restrict__ LOG,
                          const float* __restrict__ SUM,
                          const int* __restrict__ erow,
                          const int* __restrict__ ecol,
                          float* __restrict__ AGG) {
  const int t = threadIdx.x;
  const int wave = t >> 5, lane = t & 31;
  const int e = blockIdx.x * 8 + wave;
  const int r = erow[e], c = ecol[e];
#pragma unroll
  for (int h = 0; h < H_HEADS; ++h) {
    const float alpha = LOG[(size_t)h * NEDGE + e] / SUM[h];
    const float4 q4 =
        *(const float4*)(Q + (size_t)r * FINAL + h * 128 + lane * 4);
    float* dst = AGG + (size_t)c * FINAL + h * 128 + lane * 4;
    unsafeAtomicAdd(dst + 0, q4.x * alpha);
    unsafeAtomicAdd(dst + 1, q4.y * alpha);
    unsafeAtomicAdd(dst + 2, q4.z * alpha);
    unsafeAtomicAdd(dst + 3, q4.w * alpha);
  }
}

// ---------------------------------------------------------------------------
// Kernel 5: h = concat(agg)+x ; LayerNorm(512) ; SiLU ; += residual R.
// Block per node, thread handles cols t and t+256. Result overwrites R.
// ---------------------------------------------------------------------------
__global__ void k_ln_silu(const float* __restrict__ AGG,
                          const float* __restrict__ x,
                          const float* __restrict__ gamma,
                          const float* __restrict__ beta,
                          float* __restrict__ R) {
  __shared__ float s1[256], s2[256];
  const int n = blockIdx.x, t = threadIdx.x;
  const int c1 = t + 256;
  const float v0 = AGG[(size_t)n * FINAL + t]  + x[(size_t)n * D_IN + (t & 127)];
  const float v1 = AGG[(size_t)n * FINAL + c1] + x[(size_t)n * D_IN + (c1 & 127)];
  s1[t] = v0 + v1;
  s2[t] = v0 * v0 + v1 * v1;
  __syncthreads();
  for (int off = 128; off > 0; off >>= 1) {
    if (t < off) { s1[t] += s1[t + off]; s2[t] += s2[t + off]; }
    __syncthreads();
  }
  const float mu   = s1[0] * (1.0f / FINAL);
  const float var  = s2[0] * (1.0f / FINAL) - mu * mu;
  const float rstd = rsqrtf(var + 1e-5f);
  {
    const float hn = (v0 - mu) * rstd * gamma[t] + beta[t];
    const float hs = hn / (1.0f + __expf(-hn)); // hn * sigmoid(hn)
    R[(size_t)n * FINAL + t] = hs + R[(size_t)n * FINAL + t];
  }
  {
    const float hn = (v1 - mu) * rstd * gamma[c1] + beta[c1];
    const float hs = hn / (1.0f + __expf(-hn));
    R[(size_t)n * FINAL + c1] = hs + R[(size_t)n * FINAL + c1];
  }
}

// ---------------------------------------------------------------------------
// Kernel 6: out = S @ Wo^T + bo. (N,512)x(512,128). WMMA, S staged in LDS.
// ---------------------------------------------------------------------------
__global__ void k_out_gemm(const float* __restrict__ S,
                           const float* __restrict__ Wo,
                           const float* __restrict__ bo,
                           float* __restrict__ out) {
  __shared__ float ss[16 * FINAL]; // 32 KB
  const int r0 = blockIdx.x * 16;
  const int t = threadIdx.x;
#pragma unroll
  for (int j = 0; j < 8; ++j) {
    const int li = (t + j * 256) * 4;       // float index 0..8191
    const int row = li >> 9, col = li & 511;
    *(float4*)&ss[li] = *(const float4*)&S[(size_t)(r0 + row) * FINAL + col];
  }
  __syncthreads();

  const int wave = t >> 5, lane = t & 31;
  const int mn = lane & 15, half = lane >> 4;
  const int c0 = wave * 16;
  v8f acc = {};
#pragma unroll 8
  for (int k0 = 0; k0 < FINAL; k0 += 4) {
    v2f a = *(const v2f*)&ss[mn * FINAL + k0 + 2 * half];
    v2f b = *(const v2f*)&Wo[(size_t)(c0 + mn) * FINAL + k0 + 2 * half];
    acc = __builtin_amdgcn_wmma_f32_16x16x4_f32(
        false, a, false, b, (short)0, acc, false, false);
  }
  const float bv = bo[c0 + mn];
#pragma unroll
  for (int r = 0; r < 8; ++r) {
    const int m = r + 8 * half;
    out[(size_t)(r0 + m) * D_OUT_ + c0 + mn] = acc[r] + bv;
  }
}

// ---------------------------------------------------------------------------
extern "C" void kernel_launch(void* const* d_in, const int* in_sizes, int n_in,
                              void* d_out, int out_size, void* d_ws,
                              size_t ws_size, hipStream_t stream) {
  const float* x     = (const float*)d_in[0];
  const int*   ei    = (const int*)d_in[1];
  const float* Wq    = (const float*)d_in[2];
  const float* Wk    = (const float*)d_in[3];
  const float* gamma = (const float*)d_in[4];
  const float* beta  = (const float*)d_in[5];
  const float* Wr    = (const float*)d_in[6];
  const float* br    = (const float*)d_in[7];
  const float* Wo    = (const float*)d_in[8];
  const float* bo    = (const float*)d_in[9];
  float* out = (float*)d_out;

  // Workspace layout (floats). Total = 3*N*512 + H*E + 8 = 32,000,008 f = 128 MB.
  float* Q   = (float*)d_ws;                    // (N,512)
  float* K   = Q + (size_t)N_NODES * FINAL;     // (N,512) — dead after logits
  float* R   = K + (size_t)N_NODES * FINAL;     // (N,512) residual -> s
  float* LOG = R + (size_t)N_NODES * FINAL;     // (H,E) logits -> exp values
  unsigned* MAXK = (unsigned*)(LOG + (size_t)H_HEADS * NEDGE); // 4 keys
  float* SUM = (float*)(MAXK + H_HEADS);        // 4 sums
  float* AGG = K;                               // alias: reuse K region

  const int* erow = ei;
  const int* ecol = ei + NEDGE;

  hipMemsetAsync(MAXK, 0, 2 * H_HEADS * sizeof(unsigned), stream); // MAXK+SUM

  k_gemm_qkr<<<dim3(N_NODES / 16, 3), 256, 0, stream>>>(
      x, Wq, Wk, Wr, br, Q, K, R);
  k_edge_logits<<<NEDGE / 8, 256, 0, stream>>>(Q, K, erow, ecol, LOG, MAXK);
  k_exp_sum<<<NEDGE / 256, 256, 0, stream>>>(LOG, MAXK, SUM);
  hipMemsetAsync(AGG, 0, (size_t)N_NODES * FINAL * sizeof(float), stream);
  k_scatter<<<NEDGE / 8, 256, 0, stream>>>(Q, LOG, SUM, erow, ecol, AGG);
  k_ln_silu<<<N_NODES, 256, 0, stream>>>(AGG, x, gamma, beta, R);
  k_out_gemm<<<N_NODES / 16, 256, 0, stream>>>(R, Wo, bo, out);
}